// SparseUvuTensorProduct_55551107007089
// MI455X (gfx1250) — compile-verified
//
#include <hip/hip_runtime.h>
#include <math.h>

// ---------------------------------------------------------------------------
// SparseUvuTensorProduct for MI455X (gfx1250), wave32 + V_WMMA_F32_16X16X4_F32.
//
// Each path is a (B x 128) @ (128 x 128)^T fp32 GEMM (K=128) plus an
// elementwise epilogue. W (128x128 = 64KB) is staged in LDS with an XOR
// swizzle (float2 granularity) so B-fragment ds_load_b64 reads are
// bank-conflict free. One wave = one 16-row batch tile x full 128-col strip.
// ---------------------------------------------------------------------------

typedef float v2f __attribute__((ext_vector_type(2)));
typedef float v8f __attribute__((ext_vector_type(8)));

#define BATCH   32768
#define IN_DIM  2048
#define OUT_DIM 2432
#define MUL     128

__device__ __forceinline__ v8f wmma_f32(v2f a, v2f b, v8f c) {
  // D = A(16x4,f32) * B(4x16,f32) + C(16x16,f32)
  return __builtin_amdgcn_wmma_f32_16x16x4_f32(
      /*neg_a=*/false, a, /*neg_b=*/false, b,
      /*c_mod=*/(short)0, c, /*reuse_a=*/false, /*reuse_b=*/false);
}

// Stage one 128x128 fp32 W tile into LDS, XOR-swizzled on float2 slots:
// slot(u, vp) = u*64 + ((vp ^ u) & 63), vp = v>>1. Exactly 64KB, no padding.
__device__ __forceinline__ void stage_w(const float* __restrict__ w, v2f* sW) {
  const v2f* w2 = (const v2f*)w;
  const int t = threadIdx.x;
#pragma unroll
  for (int i = 0; i < 32; ++i) {
    const int idx = i * 256 + t;        // float2 slot 0..8191 (coalesced read)
    const int u   = idx >> 6;           // W row  (u index)
    const int vp  = idx & 63;           // float2 column (v>>1)
    sW[u * 64 + ((vp ^ u) & 63)] = w2[idx];
  }
}

// ---------------------------------------------------------------------------
// mode 0: out[b, OUTS+u] = d^-1/2 * sum_dp x1[b,IN1S+u*D+dp] *
//                          sum_v W[u,v] * x2[b,IN2S+v*D+dp]
// ---------------------------------------------------------------------------
template <int IN1S, int IN2S, int OUTS, int WOFF, int D>
__global__ __launch_bounds__(256) void tp_mode0_kernel(
    const float* __restrict__ x1, const float* __restrict__ x2,
    const float* __restrict__ weight, const float* __restrict__ mask,
    float* __restrict__ out) {
  __shared__ v2f sW[128 * 64];          // 64 KB
  stage_w(weight + WOFF, sW);
  __syncthreads();

  const int t    = threadIdx.x;
  const int lane = t & 31;
  const int wv   = t >> 5;
  const int col  = lane & 15;           // N index within tile / A row (M)
  const int hi   = lane >> 4;           // K-pair selector & M+8 selector
  const int m0   = (blockIdx.x * 8 + wv) * 16;
  const float inv = rsqrtf((float)D);

  v8f acc[8];
#pragma unroll
  for (int n = 0; n < 8; ++n) acc[n] = {};

  const float* x2row = x2 + (size_t)(m0 + col) * IN_DIM + IN2S;

#pragma unroll 1
  for (int dp = 0; dp < D; ++dp) {
    v8f c[8];
#pragma unroll
    for (int n = 0; n < 8; ++n) c[n] = {};

#pragma unroll 2
    for (int k = 0; k < 32; ++k) {
      const int kb = 4 * k + 2 * hi;    // K base for this lane half
      v2f a;
      a.x = x2row[(kb + 0) * D + dp];
      a.y = x2row[(kb + 1) * D + dp];
      const int vp = kb >> 1;
#pragma unroll
      for (int n = 0; n < 8; ++n) {
        const int u = n * 16 + col;
        const v2f b = sW[u * 64 + ((vp ^ u) & 63)];  // {W[u,kb], W[u,kb+1]}
        c[n] = wmma_f32(a, b, c[n]);
      }
    }
    // epilogue: acc += T .* x1 channel dp
#pragma unroll
    for (int n = 0; n < 8; ++n) {
      const int u = n * 16 + col;
#pragma unroll
      for (int r = 0; r < 8; ++r) {
        const int brow = m0 + r + 8 * hi;
        acc[n][r] += c[n][r] * x1[(size_t)brow * IN_DIM + IN1S + u * D + dp];
      }
    }
  }

#pragma unroll
  for (int n = 0; n < 8; ++n) {
    const int u = n * 16 + col;
    const float mk = mask[OUTS + u] * inv;
#pragma unroll
    for (int r = 0; r < 8; ++r) {
      const int brow = m0 + r + 8 * hi;
      out[(size_t)brow * OUT_DIM + OUTS + u] = acc[n][r] * mk;
    }
  }
}

// ---------------------------------------------------------------------------
// mode 1: mixed[b,u] = d^-1/2 * sum_v W[u,v] * x2[b,v]   (in2s == 0)
//         out[b, OUTS+u*D+dp] = x1[b, IN1S+u*D+dp] * mixed[b,u]
// ---------------------------------------------------------------------------
template <int IN1S, int OUTS, int WOFF, int D>
__global__ __launch_bounds__(256) void tp_mode1_kernel(
    const float* __restrict__ x1, const float* __restrict__ x2,
    const float* __restrict__ weight, const float* __restrict__ mask,
    float* __restrict__ out) {
  __shared__ v2f sW[128 * 64];          // 64 KB
  stage_w(weight + WOFF, sW);
  __syncthreads();

  const int t    = threadIdx.x;
  const int lane = t & 31;
  const int wv   = t >> 5;
  const int col  = lane & 15;
  const int hi   = lane >> 4;
  const int m0   = (blockIdx.x * 8 + wv) * 16;
  const float inv = rsqrtf((float)D);

  v8f c[8];
#pragma unroll
  for (int n = 0; n < 8; ++n) c[n] = {};

  const v2f* x2row = (const v2f*)(x2 + (size_t)(m0 + col) * IN_DIM);

#pragma unroll 2
  for (int k = 0; k < 32; ++k) {
    const int vp = 2 * k + hi;          // float2 K-slot
    const v2f a = x2row[vp];            // contiguous b64 load
#pragma unroll
    for (int n = 0; n < 8; ++n) {
      const int u = n * 16 + col;
      const v2f b = sW[u * 64 + ((vp ^ u) & 63)];
      c[n] = wmma_f32(a, b, c[n]);
    }
  }

#pragma unroll
  for (int n = 0; n < 8; ++n) {
    const int u = n * 16 + col;
#pragma unroll
    for (int r = 0; r < 8; ++r) {
      const int   brow  = m0 + r + 8 * hi;
      const float mixed = c[n][r] * inv;
      const float* x1p  = x1 + (size_t)brow * IN_DIM + IN1S + u * D;
      const float* mp   = mask + OUTS + u * D;
      float*       op   = out + (size_t)brow * OUT_DIM + OUTS + u * D;
#pragma unroll
      for (int dp = 0; dp < D; ++dp)
        op[dp] = x1p[dp] * mixed * mp[dp];
    }
  }
}

extern "C" void kernel_launch(void* const* d_in, const int* in_sizes, int n_in,
                              void* d_out, int out_size, void* d_ws, size_t ws_size,
                              hipStream_t stream) {
  const float* x1   = (const float*)d_in[0];
  const float* x2   = (const float*)d_in[1];
  const float* w    = (const float*)d_in[2];
  const float* mask = (const float*)d_in[3];
  float* out = (float*)d_out;

  const dim3 blk(256);
  const dim3 grd(BATCH / 128);          // 8 waves/block * 16 rows/wave

  // mode 0 paths: (in1s, in2s, outs, ws, d)
  tp_mode0_kernel<0,    0,    0,   0 * MUL * MUL, 1><<<grd, blk, 0, stream>>>(x1, x2, w, mask, out);
  tp_mode0_kernel<128,  128,  128, 1 * MUL * MUL, 3><<<grd, blk, 0, stream>>>(x1, x2, w, mask, out);
  tp_mode0_kernel<512,  512,  256, 2 * MUL * MUL, 5><<<grd, blk, 0, stream>>>(x1, x2, w, mask, out);
  tp_mode0_kernel<1152, 1152, 384, 3 * MUL * MUL, 7><<<grd, blk, 0, stream>>>(x1, x2, w, mask, out);

  // mode 1 paths: (in1s, outs, ws, d); in2s == 0
  tp_mode1_kernel<128,  512,  4 * MUL * MUL, 3><<<grd, blk, 0, stream>>>(x1, x2, w, mask, out);
  tp_mode1_kernel<512,  896,  5 * MUL * MUL, 5><<<grd, blk, 0, stream>>>(x1, x2, w, mask, out);
  tp_mode1_kernel<1152, 1536, 6 * MUL * MUL, 7><<<grd, blk, 0, stream>>>(x1, x2, w, mask, out);
}